// GRUD_60060822667646
// MI455X (gfx1250) — compile-verified
//
#include <hip/hip_runtime.h>

// GRU-D style scan: T=512 steps over [B=256, F=256] state.
// All parameter ops in the reference are element-wise (Hadamard) — there is
// no matrix multiply, so WMMA is inapplicable. Each (b,f) chain is fully
// independent: one thread per element, h in a register, serial scan over T.
// Pure streaming workload: ~537 MB traffic => ~23 us at 23.3 TB/s HBM.
// CDNA5 paths: wave32 streaming, clause-grouped global_load_b32 with
// scale_offset, speculative global_prefetch_b8 ahead of the serial chain.

#define T_STEPS 512
#define B_SZ    256
#define F_SZ    256
#define STRIDE  (B_SZ * F_SZ)   // elements between consecutive time steps
#define PF_DIST 6               // prefetch distance (time steps)

__global__ __launch_bounds__(256) void grud_scan_kernel(
    const float* __restrict__ X,      // [T,B,F]
    const float* __restrict__ Mask,   // [T,B,F]
    const float* __restrict__ Delta,  // [T,B,F]
    const float* __restrict__ x_mean,
    const float* __restrict__ w_dg_x, const float* __restrict__ b_dg_x,
    const float* __restrict__ w_dg_h, const float* __restrict__ b_dg_h,
    const float* __restrict__ w_xz,   const float* __restrict__ u_hz, const float* __restrict__ b_z,
    const float* __restrict__ w_xr,   const float* __restrict__ u_hr, const float* __restrict__ b_r,
    const float* __restrict__ w_xh,   const float* __restrict__ u_hh, const float* __restrict__ v_mh,
    const float* __restrict__ b_h,
    float* __restrict__ out_hidden,   // [B,T,H]
    float* __restrict__ out_last)     // [B,H]
{
    const int gid = blockIdx.x * blockDim.x + threadIdx.x;  // gid = b*F + f
    const int f   = gid & (F_SZ - 1);
    const int b   = gid >> 8;

    // Per-feature parameters: loaded once, coalesced (consecutive lanes -> consecutive f).
    const float xm  = x_mean[f];
    const float wdx = w_dg_x[f], bdx = b_dg_x[f];
    const float wdh = w_dg_h[f], bdh = b_dg_h[f];
    const float wxz = w_xz[f],   uhz = u_hz[f],  bz = b_z[f];
    const float wxr = w_xr[f],   uhr = u_hr[f],  br = b_r[f];
    const float wxh = w_xh[f],   uhh = u_hh[f],  vmh = v_mh[f], bh = b_h[f];

    const float* __restrict__ xp = X     + gid;
    const float* __restrict__ mp = Mask  + gid;
    const float* __restrict__ dp = Delta + gid;
    float* __restrict__ op = out_hidden + (size_t)b * (T_STEPS * F_SZ) + f;

    float h = 0.0f;

    // One GRU-D step: updates h from (x, m, d) and returns new h.
    auto step = [&](float x, float m, float d) -> float {
        // Decay gates: gamma = exp(-relu(w*d + b)) in (0,1].
        const float gx = __expf(-fmaxf(0.0f, fmaf(wdx, d, bdx)));
        const float gh = __expf(-fmaxf(0.0f, fmaf(wdh, d, bdh)));

        // Input imputation:
        //   xi = gx*x + (1-gx)*xm  = fma(gx, x-xm, xm)
        //   xc = m*x + (1-m)*xi    = fma(m, x-xi, xi)
        const float xi = fmaf(gx, x - xm, xm);
        const float xc = fmaf(m,  x - xi, xi);

        // Hidden decay.
        const float hh = gh * h;

        // z, r gates: sigmoid(a) = 1/(1+exp(-a)); saturates cleanly at +-inf.
        const float az = fmaf(wxz, xc, fmaf(uhz, hh, bz));
        const float ar = fmaf(wxr, xc, fmaf(uhr, hh, br));
        const float z  = 1.0f / (1.0f + __expf(-az));
        const float r  = 1.0f / (1.0f + __expf(-ar));

        // Candidate: tanh(p) = 1 - 2/(exp(2p)+1); exp overflow -> exactly +-1.
        const float pre = fmaf(wxh, xc, fmaf(uhh, r * hh, fmaf(vmh, m, bh)));
        const float e   = __expf(2.0f * pre);
        const float th  = 1.0f - 2.0f / (e + 1.0f);

        // h' = (1-z)*hh + z*th
        return fmaf(z, th - hh, hh);
    };

    // Software pipeline: preload step 0. Hot loop is branch-free: it loads
    // t+1 unconditionally (t runs to T-2) and the final step is peeled.
    float x = xp[0];
    float m = mp[0];
    float d = dp[0];

    #pragma unroll 4
    for (int t = 0; t < T_STEPS - 1; ++t) {
        // Speculative prefetch of step t+PF_DIST (global_prefetch_b8).
        // Past-the-end addresses are safe: speculative prefetch translation
        // failures are silently dropped (no fault, no counter).
        __builtin_prefetch(xp + (size_t)(t + PF_DIST) * STRIDE, 0, 0);
        __builtin_prefetch(mp + (size_t)(t + PF_DIST) * STRIDE, 0, 0);
        __builtin_prefetch(dp + (size_t)(t + PF_DIST) * STRIDE, 0, 0);

        // Next step's loads: independent of the serial h chain, issued early
        // so they overlap the VALU work of step t.
        const float xn = xp[(size_t)(t + 1) * STRIDE];
        const float mn = mp[(size_t)(t + 1) * STRIDE];
        const float dn = dp[(size_t)(t + 1) * STRIDE];

        h = step(x, m, d);
        op[(size_t)t * F_SZ] = h;   // coalesced store, [B,T,H] layout

        x = xn; m = mn; d = dn;     // rotate pipeline registers
    }

    // Peeled final step (no load of t+1).
    h = step(x, m, d);
    op[(size_t)(T_STEPS - 1) * F_SZ] = h;

    out_last[gid] = h;              // final hidden state [B,H]
}

extern "C" void kernel_launch(void* const* d_in, const int* in_sizes, int n_in,
                              void* d_out, int out_size, void* d_ws, size_t ws_size,
                              hipStream_t stream) {
    const float* X      = (const float*)d_in[0];
    const float* Mask   = (const float*)d_in[1];
    const float* Delta  = (const float*)d_in[2];
    const float* x_mean = (const float*)d_in[3];
    const float* w_dg_x = (const float*)d_in[4];
    const float* b_dg_x = (const float*)d_in[5];
    const float* w_dg_h = (const float*)d_in[6];
    const float* b_dg_h = (const float*)d_in[7];
    const float* w_xz   = (const float*)d_in[8];
    const float* u_hz   = (const float*)d_in[9];
    const float* b_z    = (const float*)d_in[10];
    const float* w_xr   = (const float*)d_in[11];
    const float* u_hr   = (const float*)d_in[12];
    const float* b_r    = (const float*)d_in[13];
    const float* w_xh   = (const float*)d_in[14];
    const float* u_hh   = (const float*)d_in[15];
    const float* v_mh   = (const float*)d_in[16];
    const float* b_h    = (const float*)d_in[17];

    float* out_hidden = (float*)d_out;                                  // [B,T,H]
    float* out_last   = out_hidden + (size_t)B_SZ * T_STEPS * F_SZ;     // [B,H]

    const int threads = 256;                       // 8 wave32s per block
    const int blocks  = (B_SZ * F_SZ) / threads;   // 65536 threads = 2048 waves

    grud_scan_kernel<<<blocks, threads, 0, stream>>>(
        X, Mask, Delta, x_mean,
        w_dg_x, b_dg_x, w_dg_h, b_dg_h,
        w_xz, u_hz, b_z,
        w_xr, u_hr, b_r,
        w_xh, u_hh, v_mh, b_h,
        out_hidden, out_last);
}